// TGCN_77129022702011
// MI455X (gfx1250) — compile-verified
//
#include <hip/hip_runtime.h>
#include <hip/hip_bf16.h>
#include <math.h>

typedef float v2f __attribute__((ext_vector_type(2)));
typedef float v8f __attribute__((ext_vector_type(8)));

#define DEV_INLINE __device__ __forceinline__

constexpr int kT   = 12;
constexpr int kN   = 20000;
constexpr int kE   = 320000;
constexpr int kFin = 128;
constexpr int kH   = 256;    // H_GCN == H_LSTM
constexpr int kCls = 8;

// native f32 atomic add (global_atomic_add_f32), relaxed/agent scope
DEV_INLINE void atomAddF32(float* p, float v) {
  __hip_atomic_fetch_add(p, v, __ATOMIC_RELAXED, __HIP_MEMORY_SCOPE_AGENT);
}

// ---------------------------------------------------------------- utilities
__global__ void zero_f4(float4* p, int n4) {
  int i = blockIdx.x * blockDim.x + threadIdx.x;
  if (i < n4) p[i] = make_float4(0.f, 0.f, 0.f, 0.f);
}

__global__ void deg_init(float* deg, int n) {
  int i = blockIdx.x * blockDim.x + threadIdx.x;
  if (i < n) deg[i] = 1.0f;                       // self-loop contributes 1
}

__global__ void deg_count(const int* __restrict__ dst, float* __restrict__ deg, int e) {
  int i = blockIdx.x * blockDim.x + threadIdx.x;
  if (i < e) atomAddF32(&deg[dst[i]], 1.0f);
}

__global__ void deg_rsqrt(const float* __restrict__ deg, float* __restrict__ dinv, int n) {
  int i = blockIdx.x * blockDim.x + threadIdx.x;
  if (i < n) dinv[i] = rsqrtf(deg[i]);            // deg >= 1 always
}

// ----------------------------------------------------- fp32 WMMA GEMM
// C[M x 256] = A[M x K] * B[K x 256].  Block = 256 threads (8 waves) computes a
// 32 x 256 tile: wave w -> rows (w>>2)*16, cols (w&3)*64 (4 accumulator tiles).
// B K-panels (16 x 256, 16 KB) are staged in LDS with CDNA5 async-DMA loads
// (GLOBAL_LOAD_ASYNC_TO_LDS_B128, ASYNCcnt) and consumed by
// V_WMMA_F32_16X16X4_F32.  Requires M % 32 == 0, K % 16 == 0 (grid exact, no
// guards -> no divergent barriers, EXEC all-ones for WMMA).
// A frag (16x4 f32): lanes 0-15 hold K={k,k+1}, lanes 16-31 hold K={k+2,k+3}.
// C/D (16x16 f32): VGPR r -> row r (lanes 0-15) / row r+8 (lanes 16-31).
__global__ void __launch_bounds__(256)
gemm_wmma_f32(const float* __restrict__ A,
              const float* __restrict__ B,
              float* __restrict__ C,
              int M, int K) {
  __shared__ float bpanel[16 * kH];                  // 16 K-rows x 256 cols

  const int tid  = threadIdx.x;
  const int wave = tid >> 5;
  const int lane = tid & 31;
  const int hi   = lane >> 4;                        // lane half
  const int lm   = lane & 15;

  const int msub = wave >> 2;                        // 0..1  (16-row subtile)
  const int ngrp = wave & 3;                         // 0..3  (64-col strip)
  const int row0 = blockIdx.x * 32 + msub * 16;

  const float* arow = A + (size_t)(row0 + lm) * K;
  // low 32 bits of the generic shared-pointer == LDS byte offset (ISA 10.2)
  const unsigned lds_base = (unsigned)(size_t)&bpanel[0];

  v8f acc[4] = {};

  for (int p = 0; p < K; p += 16) {
    // ---- stage B[p..p+16) x 256 into LDS: 1024 float4, one b128 per thread x4
#pragma unroll
    for (int i = 0; i < 4; ++i) {
      int f = tid + i * 256;                         // float4 index in [0,1024)
      int brow = f >> 6;
      int bcol = (f & 63) << 2;
      const float* gp = B + (size_t)(p + brow) * kH + bcol;
      unsigned lo = lds_base + (unsigned)(f << 4);
      asm volatile("global_load_async_to_lds_b128 %0, %1, off"
                   :: "v"(lo), "v"(gp) : "memory");
    }
    asm volatile("s_wait_asynccnt 0" ::: "memory");  // own DMAs complete
    __syncthreads();                                 // everyone's DMAs visible

#pragma unroll
    for (int ks = 0; ks < 4; ++ks) {
      v2f a = *(const v2f*)(arow + p + ks * 4 + (hi << 1));
      int kb = ks * 4 + (hi << 1);
#pragma unroll
      for (int nt = 0; nt < 4; ++nt) {
        int col = ngrp * 64 + nt * 16 + lm;
        v2f b;
        b.x = bpanel[kb * kH + col];
        b.y = bpanel[(kb + 1) * kH + col];
        acc[nt] = __builtin_amdgcn_wmma_f32_16x16x4_f32(false, a, false, b,
                                                        (short)0, acc[nt],
                                                        false, false);
      }
    }
    __syncthreads();                                 // before panel overwrite
  }

#pragma unroll
  for (int nt = 0; nt < 4; ++nt) {
    float* crow = C + (size_t)(row0 + hi * 8) * kH + ngrp * 64 + nt * 16 + lm;
#pragma unroll
    for (int r = 0; r < 8; ++r) crow[(size_t)r * kH] = acc[nt][r];
  }
}

// ----------------------------------------------------- edge scatter (L2-bound)
// One wave per edge: acc[dst] += xl[src] * dinv[src]*dinv[dst] over 256 feats.
__global__ void scatter_edges(const float* __restrict__ xl,
                              const int* __restrict__ src,
                              const int* __restrict__ dst,
                              const float* __restrict__ dinv,
                              float* __restrict__ acc, int e) {
  int edge = blockIdx.x * (blockDim.x >> 5) + (threadIdx.x >> 5);
  if (edge >= e) return;
  int lane = threadIdx.x & 31;
  int s = src[edge];
  int d = dst[edge];
  const float* xs = xl + (size_t)s * kH;
  __builtin_prefetch(xs + lane * 8, 0, 0);                     // global_prefetch_b8
  float norm = dinv[s] * dinv[d];
  float* ad = acc + (size_t)d * kH;
#pragma unroll
  for (int i = 0; i < 2; ++i) {
    int f = (i * 32 + lane) * 4;
    float4 v = *(const float4*)(xs + f);
    atomAddF32(ad + f + 0, v.x * norm);
    atomAddF32(ad + f + 1, v.y * norm);
    atomAddF32(ad + f + 2, v.z * norm);
    atomAddF32(ad + f + 3, v.w * norm);
  }
}

// out = relu(acc + xl * dinv^2 (self-loop) + bias)
__global__ void gcn_finalize_relu(const float* __restrict__ acc,
                                  const float* __restrict__ xl,
                                  const float* __restrict__ dinv,
                                  const float* __restrict__ bias,
                                  float* __restrict__ out) {
  int i = blockIdx.x * blockDim.x + threadIdx.x;               // over N*64 float4
  if (i >= kN * (kH / 4)) return;
  int row = i / (kH / 4);
  int c4  = i % (kH / 4);
  float di = dinv[row];
  float sw = di * di;
  float4 a = ((const float4*)acc)[i];
  float4 x = ((const float4*)xl)[i];
  float4 b = ((const float4*)bias)[c4];
  float4 r;
  r.x = fmaxf(fmaf(x.x, sw, a.x) + b.x, 0.f);
  r.y = fmaxf(fmaf(x.y, sw, a.y) + b.y, 0.f);
  r.z = fmaxf(fmaf(x.z, sw, a.z) + b.z, 0.f);
  r.w = fmaxf(fmaf(x.w, sw, a.w) + b.w, 0.f);
  ((float4*)out)[i] = r;
}

// global mean pool: 80 blocks x 256 threads, 250 rows per block, coalesced.
__global__ void mean_pool(const float* __restrict__ h, float* __restrict__ pooled_t) {
  int f  = threadIdx.x;
  int r0 = blockIdx.x * 250;
  float s = 0.f;
  for (int r = 0; r < 250; ++r) s += h[(size_t)(r0 + r) * kH + f];
  atomAddF32(&pooled_t[f], s * (1.0f / (float)kN));
}

// ----------------------------------------------------- LSTM tail (persistent)
DEV_INLINE float sigmoidf_(float x) { return 1.0f / (1.0f + expf(-x)); }

__global__ void __launch_bounds__(1024)
lstm_head(const float* __restrict__ pooled,
          const float* __restrict__ Wih0, const float* __restrict__ Whh0,
          const float* __restrict__ bih0, const float* __restrict__ bhh0,
          const float* __restrict__ Wih1, const float* __restrict__ Whh1,
          const float* __restrict__ bih1, const float* __restrict__ bhh1,
          const float* __restrict__ Wlin, const float* __restrict__ blin,
          float* __restrict__ out) {
  __shared__ float seq[kT * kH];
  __shared__ float h[kH];
  __shared__ float c[kH];
  __shared__ float gates[4 * kH];
  int tid = threadIdx.x;

  for (int i = tid; i < kT * kH; i += 1024) seq[i] = pooled[i];
  __syncthreads();

  for (int layer = 0; layer < 2; ++layer) {
    const float* Wih = layer ? Wih1 : Wih0;
    const float* Whh = layer ? Whh1 : Whh0;
    const float* bih = layer ? bih1 : bih0;
    const float* bhh = layer ? bhh1 : bhh0;
    if (tid < kH) { h[tid] = 0.f; c[tid] = 0.f; }
    __syncthreads();

    for (int t = 0; t < kT; ++t) {
      // gates[g] = x_t . Wih[g,:] + h . Whh[g,:] + bih[g] + bhh[g]
      float sum = bih[tid] + bhh[tid];
      const float* wi = Wih + (size_t)tid * kH;
      const float* wh = Whh + (size_t)tid * kH;
      const float* xt = seq + t * kH;
      for (int k = 0; k < kH; ++k) sum += xt[k] * wi[k] + h[k] * wh[k];
      gates[tid] = sum;
      __syncthreads();

      if (tid < kH) {                                // i, f, g, o gate order
        float ig = sigmoidf_(gates[tid]);
        float fg = sigmoidf_(gates[kH + tid]);
        float gg = tanhf(gates[2 * kH + tid]);
        float og = sigmoidf_(gates[3 * kH + tid]);
        float cn = fg * c[tid] + ig * gg;
        c[tid] = cn;
        float hn = og * tanhf(cn);
        h[tid] = hn;
        seq[t * kH + tid] = hn;                      // feeds next layer
      }
      __syncthreads();
    }
  }

  if (tid < kCls) {
    float s = blin[tid];
    const float* wl = Wlin + (size_t)tid * kH;
    for (int k = 0; k < kH; ++k) s += h[k] * wl[k];
    out[tid] = s;
  }
}

// ---------------------------------------------------------------- launcher
extern "C" void kernel_launch(void* const* d_in, const int* in_sizes, int n_in,
                              void* d_out, int out_size, void* d_ws, size_t ws_size,
                              hipStream_t stream) {
  (void)in_sizes; (void)n_in; (void)out_size; (void)ws_size;

  const float* x    = (const float*)d_in[0];
  const int*   ei   = (const int*)d_in[1];
  // d_in[2] = batch (all zeros, NUM_GRAPHS == 1) -> mean over all N nodes
  const float* W1   = (const float*)d_in[3];
  const float* b1   = (const float*)d_in[4];
  const float* W2   = (const float*)d_in[5];
  const float* b2   = (const float*)d_in[6];
  const float* Wih0 = (const float*)d_in[7];
  const float* Whh0 = (const float*)d_in[8];
  const float* bih0 = (const float*)d_in[9];
  const float* bhh0 = (const float*)d_in[10];
  const float* Wih1 = (const float*)d_in[11];
  const float* Whh1 = (const float*)d_in[12];
  const float* bih1 = (const float*)d_in[13];
  const float* bhh1 = (const float*)d_in[14];
  const float* Wlin = (const float*)d_in[15];
  const float* blin = (const float*)d_in[16];
  float* out = (float*)d_out;

  float* ws = (float*)d_ws;
  const size_t nf = (size_t)kN * kH;                 // 5,120,000 floats
  float* g      = ws;                                // GEMM output (xl)
  float* acc    = g + nf;                            // scatter accumulator
  float* h      = acc + nf;                          // conv output
  float* deg    = h + nf;
  float* dinv   = deg + kN;
  float* pooled = dinv + kN;                         // kT * kH

  const int gemmBlocks = kN / 32;                    // 625 (exact, 32 rows/block)
  const int scatBlocks = (kE + 7) / 8;               // 8 edges per block
  const int n4         = (int)(nf / 4);
  const int finThreads = kN * (kH / 4);

  for (int t = 0; t < kT; ++t) {
    const float* xt  = x + (size_t)t * kN * kFin;
    const int*   src = ei + (size_t)t * 2 * kE;
    const int*   dst = src + kE;

    deg_init <<<(kN + 255) / 256, 256, 0, stream>>>(deg, kN);
    deg_count<<<(kE + 255) / 256, 256, 0, stream>>>(dst, deg, kE);
    deg_rsqrt<<<(kN + 255) / 256, 256, 0, stream>>>(deg, dinv, kN);

    // ---- conv1: relu(gcn(x, W1) + b1)
    gemm_wmma_f32<<<gemmBlocks, 256, 0, stream>>>(xt, W1, g, kN, kFin);
    zero_f4<<<(n4 + 255) / 256, 256, 0, stream>>>((float4*)acc, n4);
    scatter_edges<<<scatBlocks, 256, 0, stream>>>(g, src, dst, dinv, acc, kE);
    gcn_finalize_relu<<<(finThreads + 255) / 256, 256, 0, stream>>>(acc, g, dinv, b1, h);

    // ---- conv2: relu(gcn(h, W2) + b2)
    gemm_wmma_f32<<<gemmBlocks, 256, 0, stream>>>(h, W2, g, kN, kH);
    zero_f4<<<(n4 + 255) / 256, 256, 0, stream>>>((float4*)acc, n4);
    scatter_edges<<<scatBlocks, 256, 0, stream>>>(g, src, dst, dinv, acc, kE);
    gcn_finalize_relu<<<(finThreads + 255) / 256, 256, 0, stream>>>(acc, g, dinv, b2, h);

    // ---- global mean pool -> pooled[t]
    zero_f4<<<1, 64, 0, stream>>>((float4*)(pooled + t * kH), kH / 4);
    mean_pool<<<kN / 250, 256, 0, stream>>>(h, pooled + t * kH);
  }

  lstm_head<<<1, 1024, 0, stream>>>(pooled, Wih0, Whh0, bih0, bhh0,
                                    Wih1, Whh1, bih1, bhh1, Wlin, blin, out);
}